// Stack_LSTM_74887049773143
// MI455X (gfx1250) — compile-verified
//
#include <hip/hip_runtime.h>
#include <math.h>

// ---- problem constants (from reference) ----
constexpr int Vv  = 32000;
constexpr int EMBd= 512;
constexpr int Hd  = 512;
constexpr int Sd  = 128;
constexpr int Bd  = 32;
constexpr int MSd = 104;
constexpr int MDd = 5;

constexpr int NWG = 32;    // persistent workgroups for recurrence
constexpr int TPB = 256;   // 8 wave32 waves per WG

typedef __attribute__((ext_vector_type(2))) float v2f;
typedef __attribute__((ext_vector_type(8))) float v8f;

#define HAS_ASYNC_LDS __has_builtin(__builtin_amdgcn_global_load_async_to_lds_b128)

#if HAS_ASYNC_LDS
// Exact parameter types per hipcc diagnostic: gcc-style int vector_size(16),
// AS(1) for the global source (printed "__device__"), AS(3) for the LDS dest.
typedef int v4i_vs __attribute__((vector_size(16)));
typedef __attribute__((address_space(1))) v4i_vs gv4i;
typedef __attribute__((address_space(3))) v4i_vs lv4i;
#endif

// D = A(16x4) * B(4x16) + C, fp32 WMMA (CDNA5)
__device__ __forceinline__ v8f wmma_f32(v2f a, v2f b, v8f c) {
  return __builtin_amdgcn_wmma_f32_16x16x4_f32(false, a, false, b, (short)0, c,
                                               false, false);
}

__device__ __forceinline__ float sigmoidf(float x) {
  return 1.0f / (1.0f + __expf(-x));
}

// Grid-wide sense-reversal barrier (monotonic generation counter).
__device__ __forceinline__ void grid_barrier(unsigned* cnt, unsigned* gen,
                                             unsigned* age) {
  __threadfence();
  __syncthreads();
  if (threadIdx.x == 0) {
    unsigned want = ++(*age);
    if (atomicAdd(cnt, 1u) == (unsigned)(NWG - 1)) {
      atomicExch(cnt, 0u);
      __threadfence();
      atomicExch(gen, want);
    } else {
      while (atomicAdd(gen, 0u) < want) __builtin_amdgcn_s_sleep(2);
    }
  }
  __syncthreads();
  __threadfence();
}

// One [32,2048] = [32,1024] x [1024,2048]^T gate GEMM; exactly 256 tiles,
// one 16x16 tile per wave (256 waves grid-wide). Split K-loop: first 512
// against Wih, second 512 against Whh (no per-iteration pointer select).
__device__ __forceinline__ void gates_gemm(const float* __restrict__ xh,
                                           const float* __restrict__ Wih,
                                           const float* __restrict__ Whh,
                                           const float* __restrict__ bih,
                                           const float* __restrict__ bhh,
                                           float* __restrict__ g) {
  const int lane = threadIdx.x & 31;
  const int gw   = blockIdx.x * (TPB / 32) + (threadIdx.x >> 5);  // 0..255
  const int n0   = (gw & 127) * 16;
  const int m0   = (gw >> 7) * 16;
  const int half = lane >> 4;
  const int lm   = lane & 15;

  v8f acc = {};
  const float* arow = xh + (m0 + lm) * 1024 + 2 * half;
  const float* w0   = Wih + (n0 + lm) * 512 + 2 * half;
  const float* w1   = Whh + (n0 + lm) * 512 + 2 * half;
#pragma unroll 8
  for (int k = 0; k < 512; k += 4)
    acc = wmma_f32(*(const v2f*)(arow + k), *(const v2f*)(w0 + k), acc);
#pragma unroll 8
  for (int k = 0; k < 512; k += 4)
    acc = wmma_f32(*(const v2f*)(arow + 512 + k), *(const v2f*)(w1 + k), acc);

  const int n = n0 + lm;
  const float bb = bih[n] + bhh[n];
#pragma unroll
  for (int r = 0; r < 8; ++r)
    g[(m0 + r + 8 * half) * 2048 + n] = acc[r] + bb;
}

// ---------------- persistent recurrence kernel ----------------
__global__ __launch_bounds__(TPB) void k_recurrence(
    const int* __restrict__ tok, const float* __restrict__ Emb,
    const float* __restrict__ Wih0, const float* __restrict__ Whh0,
    const float* __restrict__ bih0, const float* __restrict__ bhh0,
    const float* __restrict__ Wih1, const float* __restrict__ Whh1,
    const float* __restrict__ bih1, const float* __restrict__ bhh1,
    const float* __restrict__ Wn,  const float* __restrict__ bn,
    const float* __restrict__ Wa,  const float* __restrict__ ba,
    const float* __restrict__ Wsh, const float* __restrict__ bsh,
    float* __restrict__ Ht, float* __restrict__ hbuf, float* __restrict__ cbuf,
    float* __restrict__ stkA, float* __restrict__ stkB,
    float* __restrict__ xh0, float* __restrict__ xh1,
    float* __restrict__ g0, float* __restrict__ g1,
    unsigned* __restrict__ bar) {
  __shared__ float s_ne[Bd * MDd];
  __shared__ float s_aw[Bd * 2];
  unsigned age = 0;

  float* h0 = hbuf;
  float* h1 = hbuf + Bd * Hd;
  float* c0 = cbuf;
  float* c1 = cbuf + Bd * Hd;
  const int tid0 = blockIdx.x * TPB + threadIdx.x;

  for (int t = 0; t < Sd; ++t) {
    float* stkIn  = (t & 1) ? stkB : stkA;
    float* stkOut = (t & 1) ? stkA : stkB;

    // ---- Stage A: embedding gather + stack-top transform + h_mod ----
    for (int idx = tid0; idx < Bd * EMBd; idx += NWG * TPB) {
      int b = idx >> 9, k = idx & 511;
      xh0[b * 1024 + k] = Emb[(long)tok[t * Bd + b] * EMBd + k];
    }
    for (int idx = tid0; idx < Bd * Hd; idx += NWG * TPB) {
      int b = idx >> 9, h = idx & 511;
      float top = bsh[h];
      const float* srow = stkIn + b * (MSd * MDd);  // row 0 of stack
      const float* wr = Wsh + h * MDd;
#pragma unroll
      for (int d = 0; d < MDd; ++d) top += srow[d] * wr[d];
      xh0[b * 1024 + 512 + h] = h0[idx] + top;  // h_mod[0]
      xh1[b * 1024 + 512 + h] = h1[idx] + top;  // h_mod[1]
    }
    grid_barrier(bar, bar + 1, &age);

    // ---- Stage B: layer-0 gates ----
    gates_gemm(xh0, Wih0, Whh0, bih0, bhh0, g0);
    grid_barrier(bar, bar + 1, &age);

    // ---- Stage C: layer-0 elementwise LSTM ----
    for (int idx = tid0; idx < Bd * Hd; idx += NWG * TPB) {
      int b = idx >> 9, h = idx & 511;
      const float* gr = g0 + b * 2048;
      float i = sigmoidf(gr[h]);
      float fg = sigmoidf(gr[512 + h]);
      float gv = tanhf(gr[1024 + h]);
      float o = sigmoidf(gr[1536 + h]);
      float cn = fg * c0[idx] + i * gv;
      float hn = o * tanhf(cn);
      c0[idx] = cn;
      h0[idx] = hn;
      xh1[b * 1024 + h] = hn;  // layer-1 input
    }
    grid_barrier(bar, bar + 1, &age);

    // ---- Stage D: layer-1 gates ----
    gates_gemm(xh1, Wih1, Whh1, bih1, bhh1, g1);
    grid_barrier(bar, bar + 1, &age);

    // ---- Stage E: layer-1 elementwise LSTM, emit ht ----
    for (int idx = tid0; idx < Bd * Hd; idx += NWG * TPB) {
      int b = idx >> 9, h = idx & 511;
      const float* gr = g1 + b * 2048;
      float i = sigmoidf(gr[h]);
      float fg = sigmoidf(gr[512 + h]);
      float gv = tanhf(gr[1024 + h]);
      float o = sigmoidf(gr[1536 + h]);
      float cn = fg * c1[idx] + i * gv;
      float hn = o * tanhf(cn);
      c1[idx] = cn;
      h1[idx] = hn;
      Ht[(long)t * Bd * Hd + idx] = hn;
    }
    grid_barrier(bar, bar + 1, &age);

    // ---- Stage F: (redundant per-WG) ne/aw + this WG's stack slice ----
    const float* ht = Ht + (long)t * Bd * Hd;
    if (threadIdx.x < Bd * MDd) {
      int b = threadIdx.x / MDd, d = threadIdx.x % MDd;
      float s = bn[d];
      const float* hr = ht + b * Hd;
      const float* wr = Wn + d * Hd;
      for (int k = 0; k < Hd; ++k) s += hr[k] * wr[k];
      s_ne[threadIdx.x] = sigmoidf(s);
    } else if (threadIdx.x < Bd * MDd + Bd * 2) {
      int i2 = threadIdx.x - Bd * MDd;
      int b = i2 >> 1, j = i2 & 1;
      float s = ba[j];
      const float* hr = ht + b * Hd;
      const float* wr = Wa + j * Hd;
      for (int k = 0; k < Hd; ++k) s += hr[k] * wr[k];
      s_aw[i2] = s;
    }
    __syncthreads();
    if (threadIdx.x < Bd) {
      int b = threadIdx.x;
      float a0 = s_aw[2 * b], a1 = s_aw[2 * b + 1];
      float m = fmaxf(a0, a1);
      float e0 = __expf(a0 - m), e1 = __expf(a1 - m);
      float inv = 1.0f / (e0 + e1);
      s_aw[2 * b] = e0 * inv;
      s_aw[2 * b + 1] = e1 * inv;
    }
    __syncthreads();
    // each WG owns 520 of the 16640 stack elements
    for (int e = blockIdx.x * 520 + threadIdx.x; e < (blockIdx.x + 1) * 520;
         e += TPB) {
      int b = e / (MSd * MDd);
      int r = e % (MSd * MDd);
      int j = r / MDd, d = r % MDd;
      const float* sb = stkIn + b * MSd * MDd;
      float push = (j == 0) ? s_ne[b * MDd + d] : sb[(j - 1) * MDd + d];
      float pop  = (j == MSd - 1) ? 0.0f : sb[(j + 1) * MDd + d];
      stkOut[e] = s_aw[2 * b] * push + s_aw[2 * b + 1] * pop;
    }
    grid_barrier(bar, bar + 1, &age);
  }
}

// ---------------- big logits GEMM: out[4096,32000] = Ht * Wy^T + by --------
__global__ __launch_bounds__(256) void k_logits(const float* __restrict__ Ht,
                                                const float* __restrict__ Wy,
                                                const float* __restrict__ by,
                                                float* __restrict__ out) {
  // A tile 32x512 fp32 in LDS, XOR-swizzled on float4 (16B) chunks:
  // chunk q of row r lands at float offset r*512 + ((q ^ (r&15)) * 4).
  // A-fragment reads (16 lanes, rows 0..15, fixed q) then hit 16 distinct
  // banks 4 apart -> conflict-free, and 16B chunks stay contiguous for the
  // async global->LDS path.
  __shared__ float As[32 * 512];  // 64 KB
  const int m0 = blockIdx.y * 32;
  const int n0 = blockIdx.x * 256;

  const float* A = Ht + (long)m0 * Hd;
  for (int i = threadIdx.x; i < 32 * 128; i += 256) {  // 4096 float4 chunks
    int r = i >> 7, q = i & 127;
    int sq = q ^ (r & 15);
    const float* gsrc = A + r * Hd + q * 4;
    float* ldst = &As[r * 512 + sq * 4];
#if HAS_ASYNC_LDS
    __builtin_amdgcn_global_load_async_to_lds_b128(
        (gv4i*)(uintptr_t)gsrc,
        (lv4i*)(uintptr_t)(unsigned)(uintptr_t)ldst, 0, 0);
#else
    *(float4*)ldst = *(const float4*)gsrc;
#endif
  }
#if HAS_ASYNC_LDS
#if __has_builtin(__builtin_amdgcn_s_wait_asynccnt)
  __builtin_amdgcn_s_wait_asynccnt(0);
#else
  asm volatile("s_wait_asynccnt 0x0" ::: "memory");
#endif
#endif
  __syncthreads();

  const int lane = threadIdx.x & 31;
  const int w    = threadIdx.x >> 5;
  const int mi   = w & 1;   // M sub-tile 0/1
  const int ng   = w >> 1;  // 4 column groups of 64
  const int half = lane >> 4;
  const int lm   = lane & 15;
  const int nb   = n0 + ng * 64;

  const float* bptr[4];
#pragma unroll
  for (int s = 0; s < 4; ++s) {
    bptr[s] = Wy + (long)(nb + s * 16 + lm) * Hd + 2 * half;
    __builtin_prefetch(bptr[s], 0, 1);  // global_prefetch_b8
  }

  v8f acc[4] = {{}, {}, {}, {}};
  const int arow = (mi * 16 + lm) * 512;
#pragma unroll 4
  for (int k = 0; k < Hd; k += 4) {
    int p0 = (k >> 1) + half;                  // float2 pair index (0..255)
    int q  = p0 >> 1;                          // float4 chunk index
    int of = (p0 & 1) * 2;                     // float offset inside chunk
    v2f a = *(const v2f*)(&As[arow + ((q ^ lm) << 2) + of]);
#pragma unroll
    for (int s = 0; s < 4; ++s)
      acc[s] = wmma_f32(a, *(const v2f*)(bptr[s] + k), acc[s]);
  }

#pragma unroll
  for (int s = 0; s < 4; ++s) {
    int n = nb + s * 16 + lm;
    float bias = by[n];
#pragma unroll
    for (int r = 0; r < 8; ++r) {
      int m = m0 + mi * 16 + r + 8 * half;
      out[(long)m * Vv + n] = acc[s][r] + bias;
    }
  }
}

// ---------------- init: zero barrier, copy h0/c0/stack into ws -------------
__global__ void k_init(const float* __restrict__ h0in,
                       const float* __restrict__ c0in,
                       const float* __restrict__ stkin,
                       float* __restrict__ hbuf, float* __restrict__ cbuf,
                       float* __restrict__ stkA, unsigned* __restrict__ bar) {
  int idx = blockIdx.x * 256 + threadIdx.x;
  if (idx < 2) bar[idx] = 0u;
  const int N1 = 2 * Bd * Hd;
  for (int i = idx; i < N1; i += gridDim.x * 256) {
    hbuf[i] = h0in[i];
    cbuf[i] = c0in[i];
  }
  for (int i = idx; i < Bd * MSd * MDd; i += gridDim.x * 256)
    stkA[i] = stkin[i];
}

// ---------------- tail: final h/c/stack into output tuple tail -------------
__global__ void k_tail(const float* __restrict__ hbuf,
                       const float* __restrict__ cbuf,
                       const float* __restrict__ stk,
                       float* __restrict__ out) {
  const long base = (long)Sd * Bd * Vv;
  const int N1 = 2 * Bd * Hd;
  int idx = blockIdx.x * 256 + threadIdx.x;
  for (int i = idx; i < N1; i += gridDim.x * 256) {
    out[base + i] = hbuf[i];
    out[base + N1 + i] = cbuf[i];
  }
  for (int i = idx; i < Bd * MSd * MDd; i += gridDim.x * 256)
    out[base + 2 * N1 + i] = stk[i];
}

extern "C" void kernel_launch(void* const* d_in, const int* in_sizes, int n_in,
                              void* d_out, int out_size, void* d_ws,
                              size_t ws_size, hipStream_t stream) {
  const int*   tok   = (const int*)d_in[0];
  const float* h0in  = (const float*)d_in[1];
  const float* c0in  = (const float*)d_in[2];
  const float* stkin = (const float*)d_in[3];
  const float* Emb   = (const float*)d_in[4];
  const float* Wih0  = (const float*)d_in[5];
  const float* Whh0  = (const float*)d_in[6];
  const float* bih0  = (const float*)d_in[7];
  const float* bhh0  = (const float*)d_in[8];
  const float* Wih1  = (const float*)d_in[9];
  const float* Whh1  = (const float*)d_in[10];
  const float* bih1  = (const float*)d_in[11];
  const float* bhh1  = (const float*)d_in[12];
  const float* Wy    = (const float*)d_in[13];
  const float* by    = (const float*)d_in[14];
  const float* Wn    = (const float*)d_in[15];
  const float* bn    = (const float*)d_in[16];
  const float* Wa    = (const float*)d_in[17];
  const float* ba    = (const float*)d_in[18];
  const float* Wsh   = (const float*)d_in[19];
  const float* bsh   = (const float*)d_in[20];
  float* out = (float*)d_out;

  float* f = (float*)d_ws;
  float* Ht   = f; f += (long)Sd * Bd * Hd;   // 2,097,152
  float* hbuf = f; f += 2 * Bd * Hd;          // 32768
  float* cbuf = f; f += 2 * Bd * Hd;
  float* stkA = f; f += Bd * MSd * MDd;       // 16640
  float* stkB = f; f += Bd * MSd * MDd;
  float* xh0  = f; f += Bd * 1024;
  float* xh1  = f; f += Bd * 1024;
  float* g0   = f; f += Bd * 2048;
  float* g1   = f; f += Bd * 2048;
  unsigned* bar = (unsigned*)f;               // 2 words

  k_init<<<64, 256, 0, stream>>>(h0in, c0in, stkin, hbuf, cbuf, stkA, bar);
  k_recurrence<<<NWG, TPB, 0, stream>>>(
      tok, Emb, Wih0, Whh0, bih0, bhh0, Wih1, Whh1, bih1, bhh1, Wn, bn, Wa, ba,
      Wsh, bsh, Ht, hbuf, cbuf, stkA, stkB, xh0, xh1, g0, g1, bar);
  dim3 grid(Vv / 256, (Sd * Bd) / 32);  // 125 x 128
  k_logits<<<grid, 256, 0, stream>>>(Ht, Wy, by, out);
  k_tail<<<64, 256, 0, stream>>>(hbuf, cbuf, stkA, out);
}